// Net2_1_88081189306911
// MI455X (gfx1250) — compile-verified
//
#include <hip/hip_runtime.h>
#include <math.h>

typedef __attribute__((ext_vector_type(2))) float v2f;
typedef __attribute__((ext_vector_type(8))) float v8f;

#define NODES   100000
#define EDGES   1600000
#define HDIM    64
#define NEG_SLOPE 0.2f

// ---------------------------------------------------------------------------
// utility: float atomic max via int/uint ordering trick
// ---------------------------------------------------------------------------
__device__ __forceinline__ void atomic_max_f32(float* addr, float val) {
  if (val >= 0.0f) {
    atomicMax((int*)addr, __float_as_int(val));
  } else {
    atomicMin((unsigned int*)addr, (unsigned int)__float_as_int(val));
  }
}

__global__ void fill_f32(float* __restrict__ p, float v, int n) {
  int i = blockIdx.x * blockDim.x + threadIdx.x;
  if (i < n) p[i] = v;
}

// ---------------------------------------------------------------------------
// Y[N x 64] = X[N x K] @ W[K x 64] + bias      (fp32 WMMA 16x16x4)
// One wave per 16x16 output tile. N % 16 == 0 so EXEC is all ones.
// A layout: lane<16 -> rows 0..15 K{k0,k0+1}; lane>=16 -> K{k0+2,k0+3}
// B layout: mirrored; D: VGPR r -> row r (lanes 0-15) / row r+8 (lanes 16-31)
// ---------------------------------------------------------------------------
__global__ void gemm_bias_wmma(const float* __restrict__ X, const float* __restrict__ W,
                               const float* __restrict__ bias, float* __restrict__ Y,
                               int K, int tilesTotal) {
  int wid  = (blockIdx.x * blockDim.x + threadIdx.x) >> 5;
  if (wid >= tilesTotal) return;           // wave-uniform: whole wave exits together
  int lane = threadIdx.x & 31;
  int tm   = wid >> 2;                     // row tile (16 rows each)
  int tn   = wid & 3;                      // col tile (64/16 = 4)
  int half = lane >> 4;                    // 0 -> K pair {0,1}, 1 -> {2,3}
  int l    = lane & 15;
  int rowA = tm * 16 + l;
  int colB = tn * 16 + l;
  const float* xrow = X + (size_t)rowA * K;

  v8f c = {};
  for (int k0 = 0; k0 < K; k0 += 4) {
    int ka = k0 + 2 * half;
    v2f a, b;
    a.x = xrow[ka];
    a.y = xrow[ka + 1];
    b.x = W[(size_t)ka * HDIM + colB];
    b.y = W[(size_t)(ka + 1) * HDIM + colB];
    c = __builtin_amdgcn_wmma_f32_16x16x4_f32(false, a, false, b,
                                              (short)0, c, false, false);
  }
  float bb = bias[colB];
  int rbase = tm * 16 + half * 8;
  #pragma unroll
  for (int r = 0; r < 8; ++r) {
    Y[(size_t)(rbase + r) * HDIM + colB] = c[r] + bb;
  }
}

// ---------------------------------------------------------------------------
// Edge pass A: e = leaky_relu(xl[src] + xr[dst]) . att ; atomicMax per dst
// One wave (32 lanes) per edge; 2 channels per lane.
// ---------------------------------------------------------------------------
__global__ void edge_scores(const int* __restrict__ src, const int* __restrict__ dst,
                            const float* __restrict__ xl, const float* __restrict__ xr,
                            const float* __restrict__ att,
                            float* __restrict__ e, float* __restrict__ nmax) {
  int wid  = (blockIdx.x * blockDim.x + threadIdx.x) >> 5;
  if (wid >= EDGES) return;
  int lane = threadIdx.x & 31;
  int s = src[wid], d = dst[wid];
  float2 a = *(const float2*)(xl + (size_t)s * HDIM + lane * 2);
  float2 b = *(const float2*)(xr + (size_t)d * HDIM + lane * 2);
  float vx = a.x + b.x, vy = a.y + b.y;
  vx = vx > 0.0f ? vx : NEG_SLOPE * vx;
  vy = vy > 0.0f ? vy : NEG_SLOPE * vy;
  float2 w = *(const float2*)(att + lane * 2);
  float p = vx * w.x + vy * w.y;
  #pragma unroll
  for (int off = 16; off > 0; off >>= 1) p += __shfl_down(p, off, 32);
  if (lane == 0) {
    e[wid] = p;
    atomic_max_f32(&nmax[d], p);
  }
}

// ---------------------------------------------------------------------------
// Edge pass B: ex = exp(e - max[dst]) (in place); denom[dst] += ex
// ---------------------------------------------------------------------------
__global__ void edge_exp(const int* __restrict__ dst, float* __restrict__ e,
                         const float* __restrict__ nmax, float* __restrict__ nsum) {
  int i = blockIdx.x * blockDim.x + threadIdx.x;
  if (i >= EDGES) return;
  int d = dst[i];
  float ex = expf(e[i] - nmax[d]);
  e[i] = ex;
  atomicAdd(&nsum[d], ex);
}

// ---------------------------------------------------------------------------
// Edge pass C: acc[dst] += (ex/denom[dst]) * xl[src].  Wave per edge.
// ---------------------------------------------------------------------------
__global__ void edge_aggregate(const int* __restrict__ src, const int* __restrict__ dst,
                               const float* __restrict__ xl, const float* __restrict__ e,
                               const float* __restrict__ nsum, float* __restrict__ acc) {
  int wid  = (blockIdx.x * blockDim.x + threadIdx.x) >> 5;
  if (wid >= EDGES) return;
  int lane = threadIdx.x & 31;
  int s = src[wid], d = dst[wid];
  float alpha = e[wid] / nsum[d];
  float2 a = *(const float2*)(xl + (size_t)s * HDIM + lane * 2);
  float* p = acc + (size_t)d * HDIM + lane * 2;
  atomicAdd(p + 0, alpha * a.x);
  atomicAdd(p + 1, alpha * a.y);
}

// ---------------------------------------------------------------------------
// h = relu(acc + bias + skip), in place over acc.  total = NODES*HDIM
// ---------------------------------------------------------------------------
__global__ void node_update(float* __restrict__ acc, const float* __restrict__ skip,
                            const float* __restrict__ bias, int total) {
  int i = blockIdx.x * blockDim.x + threadIdx.x;
  if (i >= total) return;
  float v = acc[i] + skip[i] + bias[i & (HDIM - 1)];
  acc[i] = v > 0.0f ? v : 0.0f;
}

// ---------------------------------------------------------------------------
// out[n] = h[n] . W3 + b3   (wave per node)
// ---------------------------------------------------------------------------
__global__ void final_mv(const float* __restrict__ h, const float* __restrict__ W3,
                         const float* __restrict__ b3, float* __restrict__ out) {
  int wid  = (blockIdx.x * blockDim.x + threadIdx.x) >> 5;
  if (wid >= NODES) return;
  int lane = threadIdx.x & 31;
  float2 hv = *(const float2*)(h + (size_t)wid * HDIM + lane * 2);
  float2 wv = *(const float2*)(W3 + lane * 2);
  float p = hv.x * wv.x + hv.y * wv.y;
  #pragma unroll
  for (int off = 16; off > 0; off >>= 1) p += __shfl_down(p, off, 32);
  if (lane == 0) out[wid] = p + b3[0];
}

// ---------------------------------------------------------------------------
extern "C" void kernel_launch(void* const* d_in, const int* in_sizes, int n_in,
                              void* d_out, int out_size, void* d_ws, size_t ws_size,
                              hipStream_t stream) {
  const float* x    = (const float*)d_in[0];
  const int*   ei   = (const int*)d_in[1];
  const int*   src  = ei;
  const int*   dst  = ei + EDGES;
  const float* Wl1  = (const float*)d_in[2];
  const float* bl1  = (const float*)d_in[3];
  const float* Wr1  = (const float*)d_in[4];
  const float* br1  = (const float*)d_in[5];
  const float* att1 = (const float*)d_in[6];
  const float* b1   = (const float*)d_in[7];
  const float* Wd1  = (const float*)d_in[8];
  const float* bd1  = (const float*)d_in[9];
  const float* Wl2  = (const float*)d_in[10];
  const float* bl2  = (const float*)d_in[11];
  const float* Wr2  = (const float*)d_in[12];
  const float* br2  = (const float*)d_in[13];
  const float* att2 = (const float*)d_in[14];
  const float* b2   = (const float*)d_in[15];
  const float* Wd2  = (const float*)d_in[16];
  const float* bd2  = (const float*)d_in[17];
  const float* W3   = (const float*)d_in[18];
  const float* b3   = (const float*)d_in[19];
  float* out = (float*)d_out;

  const size_t NH = (size_t)NODES * HDIM;          // 6.4M floats
  float* ws   = (float*)d_ws;
  float* xl   = ws;                                 // [N,64]
  float* xr   = ws + NH;                            // [N,64]
  float* xd   = ws + 2 * NH;                        // [N,64]
  float* acc  = ws + 3 * NH;                        // [N,64] agg / h (in place)
  float* ev   = ws + 4 * NH;                        // [E]
  float* nmax = ev + EDGES;                         // [N]
  float* nsum = nmax + NODES;                       // [N]

  const int tiles        = (NODES / 16) * (HDIM / 16);        // 25000 waves
  const int gemmBlocks   = (tiles * 32 + 255) / 256;          // 3125
  const int edgeWBlocks  = (int)(((size_t)EDGES * 32 + 255) / 256); // 200000
  const int edgeTBlocks  = (EDGES + 255) / 256;               // 6250
  const int nodeVBlocks  = (int)((NH + 255) / 256);           // 25000
  const int nodeWBlocks  = (NODES * 32 + 255) / 256;          // 12500
  const int nBlocksN     = (NODES + 255) / 256;
  dim3 blk(256);

  // ---------------- Layer 1 ----------------
  gemm_bias_wmma<<<gemmBlocks, blk, 0, stream>>>(x, Wl1, bl1, xl, 128, tiles);
  gemm_bias_wmma<<<gemmBlocks, blk, 0, stream>>>(x, Wr1, br1, xr, 128, tiles);
  gemm_bias_wmma<<<gemmBlocks, blk, 0, stream>>>(x, Wd1, bd1, xd, 128, tiles);

  fill_f32<<<nodeVBlocks, blk, 0, stream>>>(acc, 0.0f, (int)NH);
  fill_f32<<<nBlocksN, blk, 0, stream>>>(nmax, -INFINITY, NODES);
  fill_f32<<<nBlocksN, blk, 0, stream>>>(nsum, 0.0f, NODES);

  edge_scores<<<edgeWBlocks, blk, 0, stream>>>(src, dst, xl, xr, att1, ev, nmax);
  edge_exp<<<edgeTBlocks, blk, 0, stream>>>(dst, ev, nmax, nsum);
  edge_aggregate<<<edgeWBlocks, blk, 0, stream>>>(src, dst, xl, ev, nsum, acc);
  node_update<<<nodeVBlocks, blk, 0, stream>>>(acc, xd, b1, (int)NH);  // acc = h1

  // ---------------- Layer 2 ----------------
  gemm_bias_wmma<<<gemmBlocks, blk, 0, stream>>>(acc, Wl2, bl2, xl, 64, tiles);
  gemm_bias_wmma<<<gemmBlocks, blk, 0, stream>>>(acc, Wr2, br2, xr, 64, tiles);
  gemm_bias_wmma<<<gemmBlocks, blk, 0, stream>>>(acc, Wd2, bd2, xd, 64, tiles);

  fill_f32<<<nodeVBlocks, blk, 0, stream>>>(acc, 0.0f, (int)NH);   // h1 consumed
  fill_f32<<<nBlocksN, blk, 0, stream>>>(nmax, -INFINITY, NODES);
  fill_f32<<<nBlocksN, blk, 0, stream>>>(nsum, 0.0f, NODES);

  edge_scores<<<edgeWBlocks, blk, 0, stream>>>(src, dst, xl, xr, att2, ev, nmax);
  edge_exp<<<edgeTBlocks, blk, 0, stream>>>(dst, ev, nmax, nsum);
  edge_aggregate<<<edgeWBlocks, blk, 0, stream>>>(src, dst, xl, ev, nsum, acc);
  node_update<<<nodeVBlocks, blk, 0, stream>>>(acc, xd, b2, (int)NH);  // acc = h2

  // ---------------- Head ----------------
  final_mv<<<nodeWBlocks, blk, 0, stream>>>(acc, W3, b3, out);
}